// GatingNetwork_85839216378508
// MI455X (gfx1250) — compile-verified
//
#include <hip/hip_runtime.h>
#include <math.h>

typedef __attribute__((ext_vector_type(2))) float v2f;
typedef __attribute__((ext_vector_type(8))) float v8f;

#define EPSF 1e-8f
#define LNEPS 1e-5f

// ---------------- helpers ----------------

__device__ __forceinline__ float wave_sum(float v) {
#pragma unroll
  for (int o = 16; o > 0; o >>= 1) v += __shfl_xor(v, o, 32);
  return v;
}

// block (256 threads) sum via wave reduce + 8-entry LDS
__device__ __forceinline__ float block_sum256(float v, float* s8, int tid) {
  v = wave_sum(v);
  __syncthreads();
  if ((tid & 31) == 0) s8[tid >> 5] = v;
  __syncthreads();
  float r = 0.f;
#pragma unroll
  for (int w = 0; w < 8; ++w) r += s8[w];
  return r;
}

// branchless insert of x into sorted-descending {t0..t4}, dropping the smallest
#define INS5(x)                                            \
  {                                                        \
    float _x = (x), _h;                                    \
    _h = fmaxf(t0, _x); _x = fminf(t0, _x); t0 = _h;       \
    _h = fmaxf(t1, _x); _x = fminf(t1, _x); t1 = _h;       \
    _h = fmaxf(t2, _x); _x = fminf(t2, _x); t2 = _h;       \
    _h = fmaxf(t3, _x); _x = fminf(t3, _x); t3 = _h;       \
    _h = fmaxf(t4, _x); _x = fminf(t4, _x); t4 = _h;       \
  }

// ---------------- kernel 1: feature extraction ----------------
// grid = B blocks, 256 threads (8 waves). One batch row per block.
// posteriors row (8 x 1000 f32 = 32KB) staged in LDS via async B128 copies.
__global__ __launch_bounds__(256) void feat_kernel(
    const float* __restrict__ post, float* __restrict__ feats) {
  __shared__ __align__(16) float sp[8000];  // p[e][c]
  __shared__ float smean[1000];             // mean over experts per class
  __shared__ float slogm[1000];             // log(mean + eps)
  __shared__ float s8[8];
  __shared__ float sfeat[60];
  __shared__ float smax[8];

  const int tid  = threadIdx.x;
  const int lane = tid & 31;
  const int wid  = tid >> 5;
  const size_t b = blockIdx.x;

  // async global->LDS staging of the whole 32KB row (16B per lane per op).
  // GVS addressing: 64-bit SGPR base + per-lane 32-bit byte offset.
  {
    const unsigned lds_base = (unsigned)(unsigned long long)(&sp[0]);
    const unsigned row_byte = (unsigned)(b * 32000u);
    for (int i = tid; i < 2000; i += 256) {
      unsigned goff = row_byte + (unsigned)i * 16u;  // byte offset from base
      unsigned loff = lds_base + (unsigned)i * 16u;  // LDS byte address
      asm volatile("global_load_async_to_lds_b128 %0, %1, %2"
                   :
                   : "v"(loff), "v"(goff), "s"(post)
                   : "memory");
    }
    asm volatile("s_wait_asynccnt 0" ::: "memory");
  }
  __syncthreads();

  // per-class cross-expert stats
  float var_acc = 0.f, ment_acc = 0.f, msq_acc = 0.f;
  for (int c = tid; c < 1000; c += 256) {
    float s = 0.f, sq = 0.f;
#pragma unroll
    for (int e = 0; e < 8; ++e) {
      float v = sp[e * 1000 + c];
      s += v; sq += v * v;
    }
    float m  = s * 0.125f;
    float lm = __logf(m + EPSF);
    smean[c] = m;
    slogm[c] = lm;
    ment_acc -= m * lm;                            // mean-dist entropy partial
    var_acc  += (sq - 8.f * m * m) * (1.f / 7.f);  // class var (ddof=1)
    msq_acc  += m * m;                             // ||mean||^2
  }
  float var_sum  = block_sum256(var_acc, s8, tid);
  float ment_sum = block_sum256(ment_acc, s8, tid);
  float msq_sum  = block_sum256(msq_acc, s8, tid);
  float mn = fmaxf(sqrtf(msq_sum), EPSF);

  // per-expert reduction: wave `wid` handles expert `wid`
  const int e = wid;
  float sent = 0.f, pp = 0.f, dotm = 0.f, klb = 0.f;
  float t0 = -1.f, t1 = -1.f, t2 = -1.f, t3 = -1.f, t4 = -1.f;
  for (int c = lane; c < 1000; c += 32) {
    float v  = sp[e * 1000 + c];
    float lv = __logf(v + EPSF);
    sent += v * lv;          // sum p*log(p+eps)  (entropy = -sent)
    pp   += v * v;           // ||p||^2
    dotm += v * smean[c];    // p . mean
    klb  += v * slogm[c];    // sum p*log(mean+eps)
    INS5(v);
  }
  sent = wave_sum(sent);
  pp   = wave_sum(pp);
  dotm = wave_sum(dotm);
  klb  = wave_sum(klb);
  // butterfly merge of per-lane top-5 lists -> global top-5 in every lane
#pragma unroll
  for (int o = 16; o > 0; o >>= 1) {
    float o0 = __shfl_xor(t0, o, 32);
    float o1 = __shfl_xor(t1, o, 32);
    float o2 = __shfl_xor(t2, o, 32);
    float o3 = __shfl_xor(t3, o, 32);
    float o4 = __shfl_xor(t4, o, 32);
    INS5(o0); INS5(o1); INS5(o2); INS5(o3); INS5(o4);
  }
  if (lane == 0) {
    float mass = t0 + t1 + t2 + t3 + t4;
    float pn   = fmaxf(sqrtf(pp), EPSF);
    sfeat[ 0 + e] = -sent;             // entropy
    sfeat[ 8 + e] = mass;              // topk_mass
    sfeat[16 + e] = 1.f - mass;        // residual
    sfeat[24 + e] = t0;                // max_prob
    sfeat[32 + e] = t0 - t1;           // top_gap
    sfeat[40 + e] = dotm / (pn * mn);  // cos
    sfeat[48 + e] = sent - klb;        // KL(p || mean)
    smax[e] = t0;
  }
  __syncthreads();
  if (tid == 0) {
    float mm = 0.f;
#pragma unroll
    for (int i = 0; i < 8; ++i) mm += smax[i];
    mm *= 0.125f;
    float sv = 0.f;
#pragma unroll
    for (int i = 0; i < 8; ++i) {
      float d = smax[i] - mm;
      sv += d * d;
    }
    sfeat[56] = ment_sum;                  // mean_ent
    sfeat[57] = var_sum * (1.f / 1000.f);  // mean_class_var
    sfeat[58] = sqrtf(sv * (1.f / 7.f));   // std_max (ddof=1)
  }
  __syncthreads();
  if (tid < 59) {
    float v = sfeat[tid];
    feats[b * 59 + tid] = fminf(fmaxf(v, -100.f), 100.f);
  }
}

// ---------------- kernel 2: MLP with fp32 WMMA ----------------

__device__ __forceinline__ v8f wmma4(v2f a, v2f b, v8f c) {
  // D = A(16x4,f32) * B(4x16,f32) + C(16x16,f32)
  return __builtin_amdgcn_wmma_f32_16x16x4_f32(
      false, a, false, b, (short)0, c, false, false);
}

__device__ __forceinline__ int imin(int a, int b) { return a < b ? a : b; }

// grid = B/16 blocks, 32 threads (1 wave). One 16-row batch tile per wave.
__global__ __launch_bounds__(32) void mlp_kernel(
    const float* __restrict__ feats,
    const float* __restrict__ W1, const float* __restrict__ b1,
    const float* __restrict__ g1, const float* __restrict__ be1,
    const float* __restrict__ W2, const float* __restrict__ b2,
    const float* __restrict__ g2, const float* __restrict__ be2,
    const float* __restrict__ W3, const float* __restrict__ b3,
    float* __restrict__ weights_out, float* __restrict__ logits_out) {
  __shared__ __align__(16) float A[16 * 64];  // feats tile, K padded 59 -> 64
  __shared__ __align__(16) float H1[16 * 256];
  __shared__ __align__(16) float H2[16 * 128];
  __shared__ float LG[16 * 8];

  const int lane    = threadIdx.x;   // 0..31
  const int mrow    = lane & 15;     // M (and N) index within tile
  const int khalf   = lane >> 4;     // selects K pair {0,1} vs {2,3} (ISA A layout)
  const int rowbase = blockIdx.x * 16;

  // stage zero-padded feats tile
  for (int i = lane; i < 16 * 64; i += 32) {
    int r = i >> 6, d = i & 63;
    A[i] = (d < 59) ? feats[(size_t)(rowbase + r) * 59 + d] : 0.f;
  }
  __syncthreads();

  // ---- layer 1: (16x64) @ (64x256) -> H1, +b1, LN, ReLU ----
  for (int t = 0; t < 16; ++t) {
    v8f acc = {0.f, 0.f, 0.f, 0.f, 0.f, 0.f, 0.f, 0.f};
    const int n = t * 16 + mrow;
#pragma unroll
    for (int kk = 0; kk < 16; ++kk) {
      const int k0 = kk * 4 + khalf * 2;
      v2f a = *(const v2f*)&A[mrow * 64 + k0];  // ds_load_b64
      // always-in-bounds load + select (no EXEC juggling)
      float w0 = W1[(size_t)imin(k0,     58) * 256 + n];
      float w1 = W1[(size_t)imin(k0 + 1, 58) * 256 + n];
      v2f bb;
      bb.x = (k0     < 59) ? w0 : 0.f;
      bb.y = (k0 + 1 < 59) ? w1 : 0.f;
      acc = wmma4(a, bb, acc);
    }
    float bias = b1[n];
#pragma unroll
    for (int r = 0; r < 8; ++r) {
      int M = r + khalf * 8;  // C/D layout: lanes 0-15 -> M=r, lanes 16-31 -> M=r+8
      H1[M * 256 + n] = acc[r] + bias;
    }
  }
  __syncthreads();
  {  // LayerNorm(256) + ReLU; lane covers (row = mrow, half = khalf)
    float s = 0.f, ss = 0.f;
    for (int j = 0; j < 128; ++j) {
      float x = H1[mrow * 256 + khalf * 128 + j];
      s += x; ss += x * x;
    }
    s  += __shfl_xor(s, 16, 32);
    ss += __shfl_xor(ss, 16, 32);
    float mu  = s * (1.f / 256.f);
    float var = ss * (1.f / 256.f) - mu * mu;
    float inv = 1.f / sqrtf(var + LNEPS);
    for (int j = 0; j < 128; ++j) {
      int col = khalf * 128 + j;
      float x = H1[mrow * 256 + col];
      float y = (x - mu) * inv * g1[col] + be1[col];
      H1[mrow * 256 + col] = fmaxf(y, 0.f);
    }
  }
  __syncthreads();

  // ---- layer 2: (16x256) @ (256x128) -> H2, +b2, LN, ReLU ----
  for (int t = 0; t < 8; ++t) {
    v8f acc = {0.f, 0.f, 0.f, 0.f, 0.f, 0.f, 0.f, 0.f};
    const int n = t * 16 + mrow;
#pragma unroll 8
    for (int kk = 0; kk < 64; ++kk) {
      const int k0 = kk * 4 + khalf * 2;
      v2f a = *(const v2f*)&H1[mrow * 256 + k0];  // ds_load_b64
      v2f bb;
      bb.x = W2[(size_t)k0 * 128 + n];
      bb.y = W2[(size_t)(k0 + 1) * 128 + n];
      acc = wmma4(a, bb, acc);
    }
    float bias = b2[n];
#pragma unroll
    for (int r = 0; r < 8; ++r) {
      int M = r + khalf * 8;
      H2[M * 128 + n] = acc[r] + bias;
    }
  }
  __syncthreads();
  {  // LayerNorm(128) + ReLU
    float s = 0.f, ss = 0.f;
    for (int j = 0; j < 64; ++j) {
      float x = H2[mrow * 128 + khalf * 64 + j];
      s += x; ss += x * x;
    }
    s  += __shfl_xor(s, 16, 32);
    ss += __shfl_xor(ss, 16, 32);
    float mu  = s * (1.f / 128.f);
    float var = ss * (1.f / 128.f) - mu * mu;
    float inv = 1.f / sqrtf(var + LNEPS);
    for (int j = 0; j < 64; ++j) {
      int col = khalf * 64 + j;
      float x = H2[mrow * 128 + col];
      float y = (x - mu) * inv * g2[col] + be2[col];
      H2[mrow * 128 + col] = fmaxf(y, 0.f);
    }
  }
  __syncthreads();

  // ---- layer 3: (16x128) @ (128x8) -> logits (N padded to 16 with zeros) ----
  {
    v8f acc = {0.f, 0.f, 0.f, 0.f, 0.f, 0.f, 0.f, 0.f};
    const int nsafe = imin(mrow, 7);
#pragma unroll 8
    for (int kk = 0; kk < 32; ++kk) {
      const int k0 = kk * 4 + khalf * 2;
      v2f a = *(const v2f*)&H2[mrow * 128 + k0];  // ds_load_b64
      float w0 = W3[(size_t)k0 * 8 + nsafe];
      float w1 = W3[(size_t)(k0 + 1) * 8 + nsafe];
      v2f bb;
      bb.x = (mrow < 8) ? w0 : 0.f;
      bb.y = (mrow < 8) ? w1 : 0.f;
      acc = wmma4(a, bb, acc);
    }
    if (mrow < 8) {
      float bias = b3[mrow];
#pragma unroll
      for (int r = 0; r < 8; ++r) {
        int M = r + khalf * 8;
        LG[M * 8 + mrow] = acc[r] + bias;
      }
    }
  }
  __syncthreads();

  // ---- softmax + write outputs (one row per lane, lanes 0..15) ----
  if (lane < 16) {
    const size_t row = (size_t)rowbase + lane;
    float l[8];
    float mx = -1e30f;
#pragma unroll
    for (int j = 0; j < 8; ++j) { l[j] = LG[lane * 8 + j]; mx = fmaxf(mx, l[j]); }
    float s = 0.f;
    float ex[8];
#pragma unroll
    for (int j = 0; j < 8; ++j) { ex[j] = __expf(l[j] - mx); s += ex[j]; }
    float invs = 1.f / s;
#pragma unroll
    for (int j = 0; j < 8; ++j) {
      weights_out[row * 8 + j] = ex[j] * invs;
      logits_out[row * 8 + j]  = l[j];
    }
  }
}

// ---------------- launch ----------------

extern "C" void kernel_launch(void* const* d_in, const int* in_sizes, int n_in,
                              void* d_out, int out_size, void* d_ws, size_t ws_size,
                              hipStream_t stream) {
  const float* post = (const float*)d_in[0];
  const float* W1   = (const float*)d_in[1];
  const float* b1   = (const float*)d_in[2];
  const float* g1   = (const float*)d_in[3];
  const float* be1  = (const float*)d_in[4];
  const float* W2   = (const float*)d_in[5];
  const float* b2   = (const float*)d_in[6];
  const float* g2   = (const float*)d_in[7];
  const float* be2  = (const float*)d_in[8];
  const float* W3   = (const float*)d_in[9];
  const float* b3   = (const float*)d_in[10];

  const int B = in_sizes[0] / 8000;  // 16384

  float* out     = (float*)d_out;
  float* weights = out;                   // (B, 8)
  float* logits  = out + (size_t)B * 8;   // (B, 8)
  float* feats   = out + (size_t)B * 16;  // (B, 59)

  feat_kernel<<<B, 256, 0, stream>>>(post, feats);
  mlp_kernel<<<B / 16, 32, 0, stream>>>(feats, W1, b1, g1, be1,
                                        W2, b2, g2, be2, W3, b3,
                                        weights, logits);
}